// LRMUCell_36309653520533
// MI455X (gfx1250) — compile-verified
//
#include <hip/hip_runtime.h>

// LMU (Legendre Memory Unit) scan for MI455X / gfx1250.
// BATCH=16, SEQ=512, IN_DIM=128, MEM_D=64, ORDER=256.

#define BATCH  16
#define SEQ    512
#define IN_DIM 128
#define MEM_D  64
#define ORDER  256
#define DN     (MEM_D * ORDER)   // 16384 state elems per batch

typedef float v2f __attribute__((ext_vector_type(2)));
typedef float v8f __attribute__((ext_vector_type(8)));

// CDNA5 fp32 WMMA: D(16x16,f32) = A(16x4,f32) * B(4x16,f32) + C
// 8-arg form: (neg_a, A, neg_b, B, c_mod, C, reuse_a, reuse_b)
__device__ __forceinline__ v8f wmma4(v2f a, v2f b, v8f c) {
  return __builtin_amdgcn_wmma_f32_16x16x4_f32(false, a, false, b,
                                               (short)0, c, false, false);
}

// ---------------------------------------------------------------------------
// Kernel 0: permute A (256x256) and memory_kernel (128x64) into the WMMA
// B-operand lane layout: for K-block kb and half h, the pair
// {M[4kb+2h][n], M[4kb+2h+1][n]} is stored contiguously so each lane issues
// one b64 load. dst[( (k>>1)*N + n )*2 + (k&1)] = src[k*N + n].
// ---------------------------------------------------------------------------
__global__ void __launch_bounds__(256)
permute_kernel(const float* __restrict__ A, const float* __restrict__ mk,
               float* __restrict__ Ap, float* __restrict__ mkp) {
  int tid = blockIdx.x * blockDim.x + threadIdx.x;
  if (tid < ORDER * ORDER) {
    int k = tid >> 8, n = tid & 255;
    Ap[(k >> 1) * (ORDER * 2) + n * 2 + (k & 1)] = A[tid];
  } else {
    int t2 = tid - ORDER * ORDER;
    if (t2 < IN_DIM * MEM_D) {
      int k = t2 >> 6, d = t2 & 63;
      mkp[(k >> 1) * (MEM_D * 2) + d * 2 + (k & 1)] = mk[t2];
    }
  }
}

// ---------------------------------------------------------------------------
// Kernel 1: u_in = x @ memory_kernel + bias.
// GEMM M=BATCH*SEQ=8192, K=128, N=64 via f32 WMMA.
// Block = 128 threads (4 waves); wave w owns N-tile w; grid = 512 M-tiles.
// ---------------------------------------------------------------------------
__global__ void __launch_bounds__(128)
uin_kernel(const float* __restrict__ x, const float* __restrict__ mkp,
           const float* __restrict__ bias, float* __restrict__ u_ws) {
  const int wave = threadIdx.x >> 5, lane = threadIdx.x & 31;
  const int hb = lane >> 4, l15 = lane & 15;
  const int mt = blockIdx.x;   // 16-row tile over B*T
  const int tN = wave;         // 16-col tile over MEM_D
  const int row = mt * 16 + l15;

  v8f acc = {};
  const float* xr = x + (size_t)row * IN_DIM;
#pragma unroll 8
  for (int kb = 0; kb < IN_DIM / 4; ++kb) {
    // A operand: lanes 0-15 hold {x[row][4kb], x[row][4kb+1]}, lanes 16-31 +2
    v2f a = *(const v2f*)(xr + kb * 4 + hb * 2);
    // B operand from pre-permuted memory_kernel: contiguous b64 per lane
    v2f bb = *(const v2f*)(mkp + ((kb * 2 + hb) * MEM_D + tN * 16 + l15) * 2);
    acc = wmma4(a, bb, acc);
  }
  const int d = tN * 16 + l15;
  const float bv = bias[d];
#pragma unroll
  for (int r = 0; r < 8; ++r) {               // D layout: M=r (+8 for hi half)
    int grow = mt * 16 + r + hb * 8;
    u_ws[(size_t)grow * MEM_D + d] = acc[r] + bv;
  }
}

// ---------------------------------------------------------------------------
// Kernel 2: the sequential LMU scan. One workgroup per batch element; the
// 64x256 f32 state m lives in LDS (row stride 260 -> bank-conflict-free
// ds_load_b64 WMMA A-operands). Per step:
//   phase 1a (VALU): partial sums of u = m_flat @ W_rec, coalesced W reads
//   phase 1b (WMMA): m @ A, 2 output tiles per wave, 64 f32 WMMAs per tile
//   reduce u, then m_new = acc + u*Bv -> LDS + streamed to out[b][t].
// ---------------------------------------------------------------------------
__global__ void __launch_bounds__(1024)
scan_kernel(const float* __restrict__ u_ws, const float* __restrict__ Wrec,
            const float* __restrict__ Ap, const float* __restrict__ Bv,
            float* __restrict__ out) {
  extern __shared__ float smem[];
  constexpr int MROW = ORDER + 4;              // 260: padded LDS row stride
  float* m_lds = smem;                         // 64 * 260 floats
  float* plds  = smem + MEM_D * MROW;          // 16 * 64 partials
  float* u_lds = plds + 16 * MEM_D;            // 64 reduced u values

  const int b    = blockIdx.x;
  const int tid  = threadIdx.x;
  const int wave = tid >> 5, lane = tid & 31;
  const int hb   = lane >> 4, l15 = lane & 15;
  // u-partial mapping: within a wave all lanes share `sub` (broadcast LDS
  // reads of m) and cover 32 consecutive d (coalesced 128B W_rec reads).
  const int d_u = tid & 63;
  const int sub = tid >> 6;                    // d' block [4sub, 4sub+4)
  // WMMA mapping: wave owns tiles (tM, tN0) and (tM, tN0+8).
  const int tM  = wave & 3;
  const int tN0 = wave >> 2;

  for (int i = tid; i < MEM_D * MROW; i += 1024) m_lds[i] = 0.f;  // m0 = 0
  __syncthreads();

  const float* uin_b = u_ws + (size_t)b * SEQ * MEM_D;
  float* out_b = out + (size_t)b * SEQ * DN;
  const int arow = (tM * 16 + l15) * MROW;     // this lane's m row base
  const int n0 = tN0 * 16 + l15;
  const int n1 = n0 + 128;
  const float bv0 = Bv[n0], bv1 = Bv[n1];

  for (int t = 0; t < SEQ; ++t) {
    // ---- phase 1a: recurrent-kernel partial dot products (VALU) ----
    float ps = 0.f;
#pragma unroll
    for (int dp = sub * 4; dp < sub * 4 + 4; ++dp) {
      const float* mrow = m_lds + dp * MROW;
      const float* wrow = Wrec + (size_t)dp * ORDER * MEM_D + d_u;
      __builtin_prefetch(wrow + ORDER * MEM_D, 0, 1);  // global_prefetch next row
#pragma unroll 8
      for (int n = 0; n < ORDER; ++n)
        ps += mrow[n] * wrow[n * MEM_D];
    }
    // ---- phase 1b: m @ A on fp32 matrix cores ----
    v8f acc0 = {}, acc1 = {};
#pragma unroll 4
    for (int kb = 0; kb < ORDER / 4; ++kb) {
      v2f a = *(const v2f*)(m_lds + arow + kb * 4 + hb * 2);   // ds_load_b64
      const float* bp = Ap + (size_t)(kb * 2 + hb) * ORDER * 2;
      v2f b0 = *(const v2f*)(bp + n0 * 2);                     // contiguous b64
      v2f b1 = *(const v2f*)(bp + n1 * 2);
      acc0 = wmma4(a, b0, acc0);
      acc1 = wmma4(a, b1, acc1);
    }
    plds[sub * MEM_D + d_u] = ps;
    __syncthreads();                 // all reads of m_t complete past here

    if (tid < MEM_D) {
      float s = uin_b[t * MEM_D + tid];
#pragma unroll
      for (int k = 0; k < 16; ++k) s += plds[k * MEM_D + tid];
      u_lds[tid] = s;
    }
    __syncthreads();

    // ---- m_{t+1} = acc + u (.) Bv : write LDS state + stream to output ----
    float* orow = out_b + (size_t)t * DN;
#pragma unroll
    for (int r = 0; r < 8; ++r) {
      int drow = tM * 16 + r + hb * 8;
      float u = u_lds[drow];
      float v0 = acc0[r] + u * bv0;
      float v1 = acc1[r] + u * bv1;
      m_lds[drow * MROW + n0] = v0;
      m_lds[drow * MROW + n1] = v1;
      orow[drow * ORDER + n0] = v0;
      orow[drow * ORDER + n1] = v1;
    }
    __syncthreads();                 // m_{t+1} visible before next iteration
  }
}

// ---------------------------------------------------------------------------
extern "C" void kernel_launch(void* const* d_in, const int* in_sizes, int n_in,
                              void* d_out, int out_size, void* d_ws, size_t ws_size,
                              hipStream_t stream) {
  (void)in_sizes; (void)n_in; (void)out_size; (void)ws_size;
  const float* x    = (const float*)d_in[0];  // (16,512,128)
  const float* mk   = (const float*)d_in[1];  // (128,64)
  const float* Wrec = (const float*)d_in[2];  // (16384,64)
  const float* bias = (const float*)d_in[3];  // (64,)
  const float* A    = (const float*)d_in[4];  // (256,256)
  const float* Bv   = (const float*)d_in[5];  // (1,256)
  float* out = (float*)d_out;                 // (16,512,16384)

  float* ws   = (float*)d_ws;
  float* u_ws = ws;                            // 8192*64  floats (2 MB)
  float* Ap   = ws + (size_t)BATCH * SEQ * MEM_D;   // 65536 floats (256 KB)
  float* mkp  = Ap + ORDER * ORDER;            // 8192 floats (32 KB)

  permute_kernel<<<(ORDER * ORDER + IN_DIM * MEM_D + 255) / 256, 256, 0, stream>>>(
      A, mk, Ap, mkp);
  uin_kernel<<<(BATCH * SEQ) / 16, 128, 0, stream>>>(x, mkp, bias, u_ws);

  const size_t lds_bytes = (size_t)(MEM_D * (ORDER + 4) + 16 * MEM_D + MEM_D) * 4;
  scan_kernel<<<BATCH, 1024, lds_bytes, stream>>>(u_ws, Wrec, Ap, Bv, out);
}